// OpNodeMessagePassing_42666205119385
// MI455X (gfx1250) — compile-verified
//
#include <hip/hip_runtime.h>
#include <hip/hip_bf16.h>
#include <stdint.h>

#define N_NODES 100000
#define FEAT 64
#define SCAN_BLOCK 1024
#define AS1 __attribute__((address_space(1)))
#define AS3 __attribute__((address_space(3)))

// ---- detect gfx1250 async global->LDS builtins (guarded; fallback = shfl path)
#if defined(__has_builtin)
#if __has_builtin(__builtin_amdgcn_global_load_async_to_lds_b32) && \
    __has_builtin(__builtin_amdgcn_s_wait_asynccnt)
#define HAVE_ASYNC_LDS 1
#endif
#endif
#ifndef HAVE_ASYNC_LDS
#define HAVE_ASYNC_LDS 0
#endif

// Hot kernel: one wave32 per node; lane owns float2 (32 lanes x 2 = 64 features).
// Edge (col,val) chunks are staged into LDS via gfx1250 async global->LDS DMA
// (ASYNCcnt path), then broadcast-read from LDS while the next gathered X row
// is prefetched with global_prefetch_b8.
__global__ __launch_bounds__(256)
void spmm_rows_kernel(const unsigned* __restrict__ offsets,
                      const unsigned* __restrict__ counts,
                      const int* __restrict__ scol,
                      const float* __restrict__ sval,
                      const float* __restrict__ X,
                      float* __restrict__ out, int n) {
    const int lane = threadIdx.x & 31;
    const int wave = threadIdx.x >> 5;
    const int node = blockIdx.x * (blockDim.x >> 5) + wave;
#if HAVE_ASYNC_LDS
    __shared__ int   s_col[8][32];
    __shared__ float s_val[8][32];
#endif
    if (node >= n) return;

    const unsigned off = offsets[node];
    const unsigned cnt = counts[node];
    const float2* __restrict__ X2 = (const float2*)X;
    float2 acc = make_float2(0.0f, 0.0f);

    for (unsigned base = 0; base < cnt; base += 32u) {
        unsigned m = cnt - base;
        if (m > 32u) m = 32u;
#if HAVE_ASYNC_LDS
        if ((unsigned)lane < m) {
            const unsigned idx = off + base + (unsigned)lane;
            // global (AS1) and LDS (AS3) pointers via integer round-trip:
            // low 32 bits of a flat LDS pointer are the LDS offset on CDNA5.
            AS1 int* gcol = (AS1 int*)(uintptr_t)(scol + idx);
            AS1 int* gval = (AS1 int*)(uintptr_t)(sval + idx);
            AS3 int* lcol = (AS3 int*)(uintptr_t)(uint32_t)(uintptr_t)&s_col[wave][lane];
            AS3 int* lval = (AS3 int*)(uintptr_t)(uint32_t)(uintptr_t)&s_val[wave][lane];
            __builtin_amdgcn_global_load_async_to_lds_b32(gcol, lcol, 0, 0);
            __builtin_amdgcn_global_load_async_to_lds_b32(gval, lval, 0, 0);
        }
        __builtin_amdgcn_s_wait_asynccnt(0);
        __builtin_amdgcn_wave_barrier();
        for (unsigned k = 0; k < m; ++k) {
            int   c = s_col[wave][k];
            float a = s_val[wave][k];
            if (k + 1u < m)  // prefetch next gathered X row (global_prefetch_b8)
                __builtin_prefetch(&X2[(size_t)s_col[wave][k + 1] * 32], 0, 0);
            float2 x = X2[(size_t)c * 32 + lane];
            acc.x = fmaf(a, x.x, acc.x);
            acc.y = fmaf(a, x.y, acc.y);
        }
        __builtin_amdgcn_wave_barrier();
#else
        int cv = 0; float av = 0.0f;
        if ((unsigned)lane < m) {
            const unsigned idx = off + base + (unsigned)lane;
            cv = scol[idx];
            av = sval[idx];
        }
        for (unsigned k = 0; k < m; ++k) {
            int   c = __shfl(cv, (int)k, 32);
            float a = __shfl(av, (int)k, 32);
            if (k + 1u < m) {
                int cn = __shfl(cv, (int)(k + 1), 32);
                __builtin_prefetch(&X2[(size_t)cn * 32], 0, 0);
            }
            float2 x = X2[(size_t)c * 32 + lane];
            acc.x = fmaf(a, x.x, acc.x);
            acc.y = fmaf(a, x.y, acc.y);
        }
#endif
    }
    float2* __restrict__ out2 = (float2*)out;
    out2[(size_t)node * 32 + lane] = acc;
}

__global__ void zero_u32_kernel(unsigned* __restrict__ p, int n) {
    for (int i = blockIdx.x * blockDim.x + threadIdx.x; i < n;
         i += gridDim.x * blockDim.x)
        p[i] = 0u;
}

// Pass 1: histogram of destination rows
__global__ void count_kernel(const int* __restrict__ row,
                             unsigned* __restrict__ counts, int E) {
    for (int e = blockIdx.x * blockDim.x + threadIdx.x; e < E;
         e += gridDim.x * blockDim.x)
        atomicAdd(&counts[row[e]], 1u);
}

// Pass 2a: per-block exclusive scan of 1024 counts; emit block totals.
__global__ __launch_bounds__(SCAN_BLOCK)
void scan_local_kernel(const unsigned* __restrict__ counts,
                       unsigned* __restrict__ excl,
                       unsigned* __restrict__ partials, int n) {
    __shared__ unsigned sh[SCAN_BLOCK];
    const int i = blockIdx.x * SCAN_BLOCK + (int)threadIdx.x;
    unsigned v = (i < n) ? counts[i] : 0u;
    sh[threadIdx.x] = v;
    __syncthreads();
    for (int off = 1; off < SCAN_BLOCK; off <<= 1) {
        unsigned t = (threadIdx.x >= (unsigned)off) ? sh[threadIdx.x - off] : 0u;
        __syncthreads();
        sh[threadIdx.x] += t;
        __syncthreads();
    }
    if (i < n) excl[i] = sh[threadIdx.x] - v;
    if (threadIdx.x == SCAN_BLOCK - 1) partials[blockIdx.x] = sh[SCAN_BLOCK - 1];
}

// Pass 2b: exclusive scan of the (<=128) block partials, single block.
__global__ __launch_bounds__(128)
void scan_partials_kernel(unsigned* __restrict__ partials, int nb) {
    __shared__ unsigned sh[128];
    unsigned v = ((int)threadIdx.x < nb) ? partials[threadIdx.x] : 0u;
    sh[threadIdx.x] = v;
    __syncthreads();
    for (int off = 1; off < 128; off <<= 1) {
        unsigned t = (threadIdx.x >= (unsigned)off) ? sh[threadIdx.x - off] : 0u;
        __syncthreads();
        sh[threadIdx.x] += t;
        __syncthreads();
    }
    if ((int)threadIdx.x < nb) partials[threadIdx.x] = sh[threadIdx.x] - v;
}

// Pass 2c: add block offsets; produce final CSR offsets and the scatter cursors.
__global__ void scan_fixup_kernel(unsigned* __restrict__ offsets,
                                  unsigned* __restrict__ cursors,
                                  const unsigned* __restrict__ partials, int n) {
    const int i = blockIdx.x * blockDim.x + threadIdx.x;
    if (i < n) {
        unsigned o = offsets[i] + partials[i >> 10];  // 1024 == SCAN_BLOCK
        offsets[i] = o;
        cursors[i] = o;
    }
}

// Pass 3: scatter edges into CSR order
__global__ void scatter_kernel(const int* __restrict__ row,
                               const int* __restrict__ col,
                               const float* __restrict__ vals,
                               unsigned* __restrict__ cursors,
                               int* __restrict__ scol,
                               float* __restrict__ sval, int E) {
    for (int e = blockIdx.x * blockDim.x + threadIdx.x; e < E;
         e += gridDim.x * blockDim.x) {
        int r = row[e];
        unsigned pos = atomicAdd(&cursors[r], 1u);
        scol[pos] = col[e];
        sval[pos] = vals[e];
    }
}

// Fallback if workspace is too small for the CSR build: wave-per-edge atomics.
__global__ __launch_bounds__(256)
void atomic_spmm_kernel(const int* __restrict__ row,
                        const int* __restrict__ col,
                        const float* __restrict__ vals,
                        const float* __restrict__ X,
                        float* __restrict__ out, int E) {
    const int lane = threadIdx.x & 31;
    const int wave = threadIdx.x >> 5;
    const int e = blockIdx.x * (blockDim.x >> 5) + wave;
    if (e >= E) return;
    const int r = row[e];
    const int c = col[e];
    const float a = vals[e];
    const float2* __restrict__ X2 = (const float2*)X;
    float2 x = X2[(size_t)c * 32 + lane];
    atomicAdd(&out[(size_t)r * 64 + 2 * lane + 0], a * x.x);
    atomicAdd(&out[(size_t)r * 64 + 2 * lane + 1], a * x.y);
}

extern "C" void kernel_launch(void* const* d_in, const int* in_sizes, int n_in,
                              void* d_out, int out_size, void* d_ws, size_t ws_size,
                              hipStream_t stream) {
    const int*   edge  = (const int*)d_in[0];    // [2, E] flattened: rows then cols
    const float* avals = (const float*)d_in[1];  // [E]
    const float* X     = (const float*)d_in[2];  // [N, 64]
    float*       out   = (float*)d_out;          // [N, 64]
    const int E = in_sizes[1];
    const int* row = edge;
    const int* col = edge + E;

    const int nScanBlocks = (N_NODES + SCAN_BLOCK - 1) / SCAN_BLOCK;  // 98 <= 128
    const size_t need = (size_t)3 * N_NODES * 4 + 128 * 4 + (size_t)E * 8;
    if (ws_size >= need) {
        char* p = (char*)d_ws;
        unsigned* counts   = (unsigned*)p; p += (size_t)N_NODES * 4;
        unsigned* offsets  = (unsigned*)p; p += (size_t)N_NODES * 4;
        unsigned* cursors  = (unsigned*)p; p += (size_t)N_NODES * 4;
        unsigned* partials = (unsigned*)p; p += 128 * 4;
        int*      scol     = (int*)p;      p += (size_t)E * 4;
        float*    sval     = (float*)p;

        zero_u32_kernel<<<(N_NODES + 255) / 256, 256, 0, stream>>>(counts, N_NODES);
        count_kernel<<<1024, 256, 0, stream>>>(row, counts, E);
        scan_local_kernel<<<nScanBlocks, SCAN_BLOCK, 0, stream>>>(counts, offsets,
                                                                  partials, N_NODES);
        scan_partials_kernel<<<1, 128, 0, stream>>>(partials, nScanBlocks);
        scan_fixup_kernel<<<(N_NODES + 255) / 256, 256, 0, stream>>>(offsets, cursors,
                                                                     partials, N_NODES);
        scatter_kernel<<<1024, 256, 0, stream>>>(row, col, avals, cursors, scol, sval, E);
        spmm_rows_kernel<<<(N_NODES + 7) / 8, 256, 0, stream>>>(offsets, counts,
                                                                scol, sval, X, out,
                                                                N_NODES);
    } else {
        zero_u32_kernel<<<(out_size + 255) / 256, 256, 0, stream>>>((unsigned*)out,
                                                                    out_size);
        atomic_spmm_kernel<<<(E + 7) / 8, 256, 0, stream>>>(row, col, avals, X, out, E);
    }
}